// MC_NCJT_RxUE_25795573579981
// MI455X (gfx1250) — compile-verified
//
#include <hip/hip_runtime.h>
#include <stdint.h>

// ---------------- problem constants (B=16, F=4096, S=14, NR=16, NTX=4) -----
#define NBF        65536          // B*F
#define NPAIR      6              // 12 data symbols -> 6 Alamouti pairs
#define PAIRS_PER_BLOCK 32
#define THREADS    64             // 2 lanes per pair: 16 pairs per wave32
#define NBLOCKS    ((NBF * NPAIR) / PAIRS_PER_BLOCK)   // 12288

// output region offsets (floats): bits0, bits1, gains0, gains1, nvar
#define OFF_BITS1  3145728u
#define OFF_G0     6291456u
#define OFF_G1     7077888u
#define OFF_NVAR   7864320u

#define LH_STRIDE  65             // float4s per pair (64 data + 1 pad)

typedef float f2 __attribute__((ext_vector_type(2)));

// ---------------- CDNA5 async global->LDS DMA (gfx1250) --------------------
__device__ __forceinline__ void async_b128(void* ldsp, const void* gp) {
  uint32_t l = (uint32_t)(uintptr_t)ldsp;          // low 32b of generic = LDS byte addr
  uint64_t a = (uint64_t)(uintptr_t)gp;
  asm volatile("global_load_async_to_lds_b128 %0, %1, off"
               :: "v"(l), "v"(a) : "memory");
}
__device__ __forceinline__ void wait_async0() {
  asm volatile("s_wait_asynccnt 0" ::: "memory");
}

// ---------------- scalar complex helpers (solve/slice only) ----------------
struct cx { float re, im; };
__device__ __forceinline__ cx cadd(cx a, cx b){ return {a.re+b.re, a.im+b.im}; }
__device__ __forceinline__ cx csub(cx a, cx b){ return {a.re-b.re, a.im-b.im}; }
__device__ __forceinline__ cx cmul(cx a, cx b){ return {a.re*b.re - a.im*b.im, a.re*b.im + a.im*b.re}; }
__device__ __forceinline__ cx cjmul(cx a, cx b){ /* conj(a)*b */ return {a.re*b.re + a.im*b.im, a.re*b.im - a.im*b.re}; }
__device__ __forceinline__ cx cscale(cx a, float s){ return {a.re*s, a.im*s}; }
__device__ __forceinline__ float cabs2(cx a){ return a.re*a.re + a.im*a.im; }

__device__ __forceinline__ f2 lo2(const float4& v){ f2 r; r.x = v.x; r.y = v.y; return r; }
__device__ __forceinline__ f2 hi2(const float4& v){ f2 r; r.x = v.z; r.y = v.w; return r; }
__device__ __forceinline__ float comp(const float4& v, int k) {
  return k==0 ? v.x : k==1 ? v.y : k==2 ? v.z : v.w;   // folds under #pragma unroll
}
__device__ __forceinline__ float comp8(const float4& a, const float4& b, int k) {
  return k < 4 ? comp(a, k) : comp(b, k - 4);
}

// 16-QAM Gray slicer with scale-adjusted threshold (bits are scale-invariant;
// emitted point is always the true constellation value)
struct SymDec { float b0, b1, b2, b3; cx v; };
__device__ __forceinline__ SymDec slice_qam16(cx y, float thr) {
  const float A1 = 0.316227766016837933f;  // 1/sqrt(10)
  const float A3 = 0.948683298050513800f;  // 3/sqrt(10)
  SymDec s;
  bool nre = y.re < 0.0f,       nim = y.im < 0.0f;
  bool ore = fabsf(y.re) > thr, oim = fabsf(y.im) > thr;  // strict >: tie -> bit 0
  s.b0 = nre ? 1.f : 0.f;  s.b1 = nim ? 1.f : 0.f;
  s.b2 = ore ? 1.f : 0.f;  s.b3 = oim ? 1.f : 0.f;
  float mr = ore ? A3 : A1, mi = oim ? A3 : A1;
  s.v.re = nre ? -mr : mr;  s.v.im = nim ? -mi : mi;
  return s;
}

// pairwise sum across the 2 lanes sharing one pair: ds_swizzle SWAPX1
__device__ __forceinline__ float red2(float v) {
  return v + __int_as_float(__builtin_amdgcn_ds_swizzle(__float_as_int(v), 0x041F));
}

// ---------------------------------------------------------------------------
__global__ __launch_bounds__(THREADS)
void mc_ncjt_rxue_kernel(const float* __restrict__ ry_re, const float* __restrict__ ry_im,
                         const float* __restrict__ h_re,  const float* __restrict__ h_im,
                         float* __restrict__ out)
{
  // DMA landing pad: per pair, 4 rows (sc = sym*2+comp) of 16 float4s, +1 pad
  __shared__ float4 lh[PAIRS_PER_BLOCK * LH_STRIDE];   // 33280 B

  const int tid = threadIdx.x;
  const int rb  = tid & 1;                 // antenna octet: r = 8*rb + k
  const int lp  = tid >> 1;                // local pair 0..31
  const int g   = blockIdx.x * PAIRS_PER_BLOCK + lp;
  const int bf  = g / NPAIR;
  const int p   = g - bf * NPAIR;
  const int se  = 2*p + (p > 0) + (p > 4); // DATA_SYMS[2p]
  const int so  = se + 1;                  // DATA_SYMS[2p+1]

  const size_t eb = ((size_t)bf * 14 + se) * 16;
  const size_t ob = ((size_t)bf * 14 + so) * 16;
  const int lr0 = 8 * rb;
  const int lb  = lp * LH_STRIDE;

  // ---- stage h via async DMA (each lane fetches exactly what it will read)
  const float4* h4r = (const float4*)h_re;
  const float4* h4i = (const float4*)h_im;
  #pragma unroll
  for (int k = 0; k < 8; ++k) {
    async_b128(&lh[lb + 0*16 + lr0 + k], h4r + eb + lr0 + k);
    async_b128(&lh[lb + 1*16 + lr0 + k], h4i + eb + lr0 + k);
    async_b128(&lh[lb + 2*16 + lr0 + k], h4r + ob + lr0 + k);
    async_b128(&lh[lb + 3*16 + lr0 + k], h4i + ob + lr0 + k);
  }

  // ---- ry: this lane's 8 antennas are contiguous -> coalesced float4 loads
  const float4 R1r0 = *(const float4*)(ry_re + eb + lr0);
  const float4 R1r1 = *(const float4*)(ry_re + eb + lr0 + 4);
  const float4 R1i0 = *(const float4*)(ry_im + eb + lr0);
  const float4 R1i1 = *(const float4*)(ry_im + eb + lr0 + 4);
  const float4 R2r0 = *(const float4*)(ry_re + ob + lr0);
  const float4 R2r1 = *(const float4*)(ry_re + ob + lr0 + 4);
  const float4 R2i0 = *(const float4*)(ry_im + ob + lr0);
  const float4 R2i1 = *(const float4*)(ry_im + ob + lr0 + 4);

  wait_async0();   // wave-private data: no block barrier needed

  // ---- pass 1 (packed f32, planar/SoA over antenna pairs) -----------------
  // Unscaled channel sums ha' = 2*ha: P'=4P, C'=4C, D'=4D, u'=2u; the ZF
  // solve then yields s/2, sliced with halved thresholds (scale-invariant).
  f2 Pv = {0,0}, Qv = {0,0};
  f2 Crev = {0,0}, CimA = {0,0}, CimB = {0,0};
  f2 Drev = {0,0}, DimA = {0,0}, DimB = {0,0};
  f2 ua = {0,0}, ub = {0,0}, uc = {0,0}, ud = {0,0};    // u0/u1 (ants 0,1)
  f2 ue = {0,0}, uf = {0,0}, ug = {0,0}, uh = {0,0};    // u2/u3 (ants 2,3)
  #pragma unroll
  for (int k = 0; k < 8; ++k) {
    float4 er = lh[lb + 0*16 + lr0 + k];
    float4 ei = lh[lb + 1*16 + lr0 + k];
    float4 pr = lh[lb + 2*16 + lr0 + k];
    float4 pi = lh[lb + 3*16 + lr0 + k];
    f2 haR01 = lo2(er) + lo2(pr), haI01 = lo2(ei) + lo2(pi);   // 2x averaged
    f2 haR23 = hi2(er) + hi2(pr), haI23 = hi2(ei) + hi2(pi);
    float r1R = comp8(R1r0,R1r1,k), r1I = comp8(R1i0,R1i1,k);
    float r2R = comp8(R2r0,R2r1,k), r2I = comp8(R2i0,R2i1,k);

    Pv += haR01*haR01 + haI01*haI01;
    Qv += haR23*haR23 + haI23*haI23;
    Crev += haR01*haR23    + haI01*haI23;      // C.re = hsum
    CimA += haR01*haI23;   CimB += haI01*haR23;
    Drev += haR01*haR23.yx + haI01*haI23.yx;   // D.re = .x - .y
    DimA += haR01*haI23.yx; DimB += haI01*haR23.yx;
    // u0 = sum conj(ha0)r1 + ha1*conj(r2); u1 = sum conj(ha1)r1 - ha0*conj(r2)
    ua += haR01*r1R + haI01*r1I;
    ub += haR01*r2R + haI01*r2I;
    uc += haR01*r1I - haI01*r1R;
    ud += haI01*r2R - haR01*r2I;
    ue += haR23*r1R + haI23*r1I;
    uf += haR23*r2R + haI23*r2I;
    ug += haR23*r1I - haI23*r1R;
    uh += haI23*r2R - haR23*r2I;
  }
  // horizontal combines, then 2-lane exchange on 14 scalars (all primed)
  float P = red2(Pv.x + Pv.y);
  float Q = red2(Qv.x + Qv.y);
  cx C { red2(Crev.x + Crev.y), red2(CimA.x - CimA.y - CimB.x + CimB.y) };
  cx D { red2(Drev.x - Drev.y), red2((DimA.x + DimA.y) - (DimB.x + DimB.y)) };
  cx u0 { red2(ua.x + ub.y), red2(uc.x + ud.y) };
  cx u1 { red2(ua.y - ub.x), red2(uc.y - ud.x) };
  cx u2 { red2(ue.x + uf.y), red2(ug.x + uh.y) };
  cx u3 { red2(ue.y - uf.x), red2(ug.y - uh.x) };

  // ---- closed-form ZF solve (Schur: M^H M = E*I); result = s_true/2 -------
  float E = cabs2(C) + cabs2(D);
  float invdet = 1.0f / (P * Q - E);
  float invP   = 1.0f / P;
  cx t;
  t = cscale(u2, P); t = csub(t, cjmul(C, u0)); t = cadd(t, cmul(D, u1));
  cx sB0 = cscale(t, invdet);
  t = cscale(u3, P); t = csub(t, cjmul(D, u0)); t = csub(t, cmul(C, u1));
  cx sB1 = cscale(t, invdet);
  t = csub(u0, cmul(C, sB0)); t = csub(t, cmul(D, sB1));
  cx sA0 = cscale(t, invP);
  t = cadd(u1, cjmul(D, sB0)); t = csub(t, cjmul(C, sB1));
  cx sA1 = cscale(t, invP);

  const bool c0 = (P >= Q);
  const float THALF = 0.316227766016837933f;   // (2/sqrt(10)) / 2
  SymDec xe = slice_qam16(c0 ? sA0 : sB0, THALF);   // better stream, even sym
  SymDec xo = slice_qam16(c0 ? sA1 : sB1, THALF);   // better stream, odd sym

  // uniform per pair: sliced symbols + Alamouti re-encode, packed over (even,odd)
  f2 xR, xI, bxR, bxI;
  xR.x  =  xe.v.re;  xR.y  =  xo.v.re;
  xI.x  =  xe.v.im;  xI.y  =  xo.v.im;
  bxR.x = -xo.v.re;  bxR.y =  xe.v.re;       // (-conj(xo), conj(xe)).re
  bxI.x =  xo.v.im;  bxI.y = -xe.v.im;       // (-conj(xo), conj(xe)).im

  // ---- pass 2: SIC residual + Alamouti combine of worse stream ------------
  // better/worse antenna halves picked by *address* (f2 slices of LDS rows)
  const f2* lh2 = (const f2*)lh;
  const int base2 = 2 * (lb + lr0);
  const int bsel  = c0 ? 0 : 1;
  const int bb2 = base2 + bsel, wb2 = base2 + (1 - bsel);
  f2 s0rev = {0,0}, s0imA = {0,0}, s0imB = {0,0};
  f2 s1rev = {0,0}, s1imA = {0,0}, s1imB = {0,0};
  #pragma unroll
  for (int k = 0; k < 8; ++k) {
    f2 cbeR = lh2[bb2 + 2*(0*16 + k)];     // better @ even sym
    f2 cbeI = lh2[bb2 + 2*(1*16 + k)];
    f2 cboR = lh2[bb2 + 2*(2*16 + k)];     // better @ odd sym
    f2 cboI = lh2[bb2 + 2*(3*16 + k)];
    f2 gwR  = lh2[wb2 + 2*(0*16 + k)] + lh2[wb2 + 2*(2*16 + k)];  // worse, 2x avg
    f2 gwI  = lh2[wb2 + 2*(1*16 + k)] + lh2[wb2 + 2*(3*16 + k)];
    float r1R = comp8(R1r0,R1r1,k), r1I = comp8(R1i0,R1i1,k);
    float r2R = comp8(R2r0,R2r1,k), r2I = comp8(R2i0,R2i1,k);

    f2 teR = cbeR*xR  - cbeI*xI;   // eff_e.re = hsum
    f2 teI = cbeR*xI  + cbeI*xR;   // eff_e.im = hsum
    f2 toR = cboR*bxR - cboI*bxI;  // eff_o.re = hsum
    f2 toI = cboR*bxI + cboI*bxR;  // eff_o.im = hsum
    f2 rnR, rnI;                   // (rn1, rn2) residuals, packed
    rnR.x = r1R - teR.x - teR.y;   rnR.y = r2R - toR.x - toR.y;
    rnI.x = r1I - teI.x - teI.y;   rnI.y = r2I - toI.x - toI.y;

    // s0 = sum conj(gw0)rn1 + gw1*conj(rn2); s1 = sum conj(gw1)rn1 - gw0*conj(rn2)
    s0rev += gwR*rnR    + gwI*rnI;
    s0imA += gwR*rnI;    s0imB += gwI*rnR;
    s1rev += gwR.yx*rnR + gwI.yx*rnI;
    s1imA += gwR.yx*rnI; s1imB += gwI.yx*rnR;
  }
  float s0re = red2(s0rev.x + s0rev.y);
  float s0im = red2(s0imA.x - s0imA.y - s0imB.x + s0imB.y);
  float s1re = red2(s1rev.x - s1rev.y);
  float s1im = red2((s1imA.x + s1imA.y) - (s1imB.x + s1imB.y));

  // s' = 2*s_true, gain_true = (c0?Q':P')/4  =>  new_y = 2*s'/(c0?Q':P')
  const float TFULL = 0.632455532033675866f;
  float invg = 2.0f / (c0 ? Q : P);
  SymDec ne = slice_qam16(cx{ s0re*invg, s0im*invg }, TFULL);
  SymDec no = slice_qam16(cx{ s1re*invg, s1im*invg }, TFULL);

  // ---- outputs (lane rb==0 of each pair) ----------------------------------
  if (rb == 0) {
    SymDec s0e = c0 ? xe : ne, s0o = c0 ? xo : no;   // stream-0 final symbols
    SymDec s1e = c0 ? ne : xe, s1o = c0 ? no : xo;   // stream-1 final symbols
    size_t bb = (size_t)bf * 48 + (size_t)p * 8;
    *(float4*)(out + bb)                  = make_float4(s0e.b0, s0e.b1, s0e.b2, s0e.b3);
    *(float4*)(out + bb + 4)              = make_float4(s0o.b0, s0o.b1, s0o.b2, s0o.b3);
    *(float4*)(out + OFF_BITS1 + bb)      = make_float4(s1e.b0, s1e.b1, s1e.b2, s1e.b3);
    *(float4*)(out + OFF_BITS1 + bb + 4)  = make_float4(s1o.b0, s1o.b1, s1o.b2, s1o.b3);
    size_t gb = (size_t)bf * 12 + (size_t)p * 2;
    *(float2*)(out + OFF_G0 + gb) = make_float2(0.25f*P, 0.25f*P);  // true gains
    *(float2*)(out + OFF_G1 + gb) = make_float2(0.25f*Q, 0.25f*Q);
    if (g == 0) out[OFF_NVAR] = 0.4f;
  }
}

extern "C" void kernel_launch(void* const* d_in, const int* in_sizes, int n_in,
                              void* d_out, int out_size, void* d_ws, size_t ws_size,
                              hipStream_t stream) {
  const float* ry_re = (const float*)d_in[0];
  const float* ry_im = (const float*)d_in[1];
  const float* h_re  = (const float*)d_in[2];
  const float* h_im  = (const float*)d_in[3];
  mc_ncjt_rxue_kernel<<<dim3(NBLOCKS), dim3(THREADS), 0, stream>>>(
      ry_re, ry_im, h_re, h_im, (float*)d_out);
}